// GAT_body_59846074302528
// MI455X (gfx1250) — compile-verified
//
#include <hip/hip_runtime.h>
#include <cstdint>
#include <cstddef>

// ---------------- problem constants (from reference) ----------------
#define N_NODES 50000
#define N_EDGES 800000
#define E_TOT   (N_EDGES + N_NODES)   // edges + self loops = 850000
#define F_IN    256
#define H1      4                      // layer-1 heads
#define C1      64                     // layer-1 out channels per head
#define C2      64                     // layer-2 out channels (1 head)
#define NEG_SLOPE 0.2f
#define EPS_F 1e-16f

typedef float v2f __attribute__((ext_vector_type(2)));
typedef float v8f __attribute__((ext_vector_type(8)));

// ---------------- utility kernels ----------------
__global__ void fill_f32(float* __restrict__ p, float v, size_t n) {
  size_t i = (size_t)blockIdx.x * blockDim.x + threadIdx.x;
  if (i < n) p[i] = v;
}

// ---------------- fp32 WMMA GEMM: C[M,N] = A[M,K] x B[K,N], row-major ----------------
// One wave computes a 16x64 strip (4 adjacent 16x16 tiles) so each A fragment
// feeds 4 V_WMMA_F32_16X16X4_F32 ops: 4x less A traffic, 2x arithmetic intensity.
// Requires N % 64 == 0, M % 16 == 0, K % 4 == 0.
// A 16x4 layout : lane l holds M = l%16, K = (l/16)*2 + {0,1} in {v0,v1}
// B 4x16 layout : lane l holds N = l%16, K = (l/16)*2 + {0,1} in {v0,v1}
// C/D 16x16     : vgpr r, lane l -> M = r + (l/16)*8, N = l%16
__global__ void gemm_wmma_f32(const float* __restrict__ A, const float* __restrict__ B,
                              float* __restrict__ C, int M, int N, int K) {
  int wave = (int)((blockIdx.x * blockDim.x + threadIdx.x) >> 5);
  int lane = threadIdx.x & 31;
  int quadsN = N >> 6;                 // 64-wide column strips
  int tm = wave / quadsN;
  int tq = wave % quadsN;
  if (tm * 16 >= M) return;            // wave-uniform exit: EXEC stays all-ones for WMMA
  int hi  = lane >> 4;                 // 0 or 1
  int l16 = lane & 15;
  int kb  = hi * 2;
  const float* arow = A + (size_t)(tm * 16 + l16) * K;
  const float* bbase = B + (size_t)tq * 64 + l16;
  v8f acc0 = {}, acc1 = {}, acc2 = {}, acc3 = {};
  for (int k = 0; k < K; k += 4) {
    v2f a;
    a.x = arow[k + kb];
    a.y = arow[k + kb + 1];
    const float* brow0 = bbase + (size_t)(k + kb) * N;
    const float* brow1 = bbase + (size_t)(k + kb + 1) * N;
    v2f b0, b1, b2, b3;
    b0.x = brow0[0];  b0.y = brow1[0];
    b1.x = brow0[16]; b1.y = brow1[16];
    b2.x = brow0[32]; b2.y = brow1[32];
    b3.x = brow0[48]; b3.y = brow1[48];
    acc0 = __builtin_amdgcn_wmma_f32_16x16x4_f32(false, a, false, b0, (short)0, acc0, false, false);
    acc1 = __builtin_amdgcn_wmma_f32_16x16x4_f32(false, a, false, b1, (short)0, acc1, false, false);
    acc2 = __builtin_amdgcn_wmma_f32_16x16x4_f32(false, a, false, b2, (short)0, acc2, false, false);
    acc3 = __builtin_amdgcn_wmma_f32_16x16x4_f32(false, a, false, b3, (short)0, acc3, false, false);
  }
  float* crow = C + (size_t)(tm * 16 + hi * 8) * N + (size_t)tq * 64 + l16;
#pragma unroll
  for (int r = 0; r < 8; ++r) {
    crow[(size_t)r * N +  0] = acc0[r];
    crow[(size_t)r * N + 16] = acc1[r];
    crow[(size_t)r * N + 32] = acc2[r];
    crow[(size_t)r * N + 48] = acc3[r];
  }
}

// ---------------- attention coefficients: alpha_s/alpha_d [N*heads] ----------------
// h layout [n][heads][C] flattened -> row i = n*heads + hh starts at h + i*C
__global__ void attn_coef(const float* __restrict__ h, const float* __restrict__ a_src,
                          const float* __restrict__ a_dst, float* __restrict__ as,
                          float* __restrict__ ad, int heads, int C) {
  int i = blockIdx.x * blockDim.x + threadIdx.x;
  if (i >= N_NODES * heads) return;
  int hh = i % heads;
  const float* row = h + (size_t)i * C;
  const float* pa = a_src + (size_t)hh * C;
  const float* pb = a_dst + (size_t)hh * C;
  float s = 0.f, d = 0.f;
#pragma unroll 4
  for (int c = 0; c < C; ++c) { float v = row[c]; s += v * pa[c]; d += v * pb[c]; }
  as[i] = s; ad[i] = d;
}

// ---------------- edge helpers ----------------
__device__ __forceinline__ void edge_sd(const long long* __restrict__ ei, int e,
                                        int& src, int& dst) {
  if (e < N_EDGES) { src = (int)ei[e]; dst = (int)ei[N_EDGES + e]; }
  else             { src = dst = e - N_EDGES; }     // self loop
}

__device__ __forceinline__ float lrelu(float x) { return x > 0.f ? x : NEG_SLOPE * x; }

// float atomic max via signed-max / unsigned-min trick (init must be -inf)
__device__ __forceinline__ void atomicMaxF(float* addr, float v) {
  if (v >= 0.f) atomicMax((int*)addr, __float_as_int(v));
  else          atomicMin((unsigned int*)addr, __float_as_uint(v));
}

// pass 1: segment max of leaky-relu'd logits per (dst, head)
__global__ void edge_max_k(const long long* __restrict__ ei, const float* __restrict__ as,
                           const float* __restrict__ ad, float* __restrict__ m, int H) {
  int i = blockIdx.x * blockDim.x + threadIdx.x;
  if (i >= E_TOT * H) return;
  int e = i / H, h = i - e * H;
  int src, dst; edge_sd(ei, e, src, dst);
  float ev = lrelu(as[src * H + h] + ad[dst * H + h]);
  atomicMaxF(&m[dst * H + h], ev);
}

// pass 2: ex = exp(e - m[dst]); denom[dst] += ex
__global__ void edge_exp_k(const long long* __restrict__ ei, const float* __restrict__ as,
                           const float* __restrict__ ad, const float* __restrict__ m,
                           float* __restrict__ ex, float* __restrict__ denom, int H) {
  int i = blockIdx.x * blockDim.x + threadIdx.x;
  if (i >= E_TOT * H) return;
  int e = i / H, h = i - e * H;
  int src, dst; edge_sd(ei, e, src, dst);
  float mv = m[dst * H + h];
  if (!(mv > -1e30f)) mv = 0.f;        // mimic where(isfinite(m), m, 0)
  float ev = lrelu(as[src * H + h] + ad[dst * H + h]);
  float x = expf(ev - mv);
  ex[i] = x;
  atomicAdd(&denom[dst * H + h], x);
}

// pass 3 (layer 1): block per edge, 256 threads = 4 heads x 64 channels
__global__ void edge_aggr1(const long long* __restrict__ ei, const float* __restrict__ h1,
                           const float* __restrict__ ex, const float* __restrict__ denom,
                           float* __restrict__ out) {
  int e = blockIdx.x;
  int t = threadIdx.x;                 // 0..255  (h = t>>6, c = t&63)
  int src, dst; edge_sd(ei, e, src, dst);
  int h = t >> 6;
  float alpha = ex[(size_t)e * H1 + h] / (denom[dst * H1 + h] + EPS_F);
  float v = h1[(size_t)src * F_IN + t] * alpha;
  atomicAdd(&out[(size_t)dst * F_IN + t], v);
}

// pass 3 (layer 2, 1 head): 64 channels per edge, 4 edges per 256-thread block
__global__ void edge_aggr2(const long long* __restrict__ ei, const float* __restrict__ h2,
                           const float* __restrict__ ex, const float* __restrict__ denom,
                           float* __restrict__ out) {
  int idx = blockIdx.x * 256 + threadIdx.x;
  int e = idx >> 6;
  if (e >= E_TOT) return;
  int c = idx & 63;
  int src, dst; edge_sd(ei, e, src, dst);
  float alpha = ex[e] / (denom[dst] + EPS_F);
  float v = h2[(size_t)src * C2 + c] * alpha;
  atomicAdd(&out[(size_t)dst * C2 + c], v);
}

// out[i] = elu(out[i] + b[i % cols])
__global__ void bias_elu(float* __restrict__ out, const float* __restrict__ b,
                         size_t n, int cols) {
  size_t i = (size_t)blockIdx.x * blockDim.x + threadIdx.x;
  if (i >= n) return;
  float x = out[i] + b[i % cols];
  out[i] = x > 0.f ? x : expm1f(x);
}

__global__ void bias_add(float* __restrict__ out, const float* __restrict__ b,
                         size_t n, int cols) {
  size_t i = (size_t)blockIdx.x * blockDim.x + threadIdx.x;
  if (i >= n) return;
  out[i] += b[i % cols];
}

// ---------------- driver ----------------
static inline int cdiv(long long a, long long b) { return (int)((a + b - 1) / b); }

extern "C" void kernel_launch(void* const* d_in, const int* in_sizes, int n_in,
                              void* d_out, int out_size, void* d_ws, size_t ws_size,
                              hipStream_t stream) {
  const float*     x    = (const float*)d_in[0];
  const long long* ei   = (const long long*)d_in[1];   // int64 [2, 800000]
  const float*     W1   = (const float*)d_in[2];
  const float*     aS1  = (const float*)d_in[3];
  const float*     aD1  = (const float*)d_in[4];
  const float*     b1   = (const float*)d_in[5];
  const float*     W2   = (const float*)d_in[6];
  const float*     aS2  = (const float*)d_in[7];
  const float*     aD2  = (const float*)d_in[8];
  const float*     b2   = (const float*)d_in[9];
  float*           out  = (float*)d_out;

  // workspace layout (floats)
  float* ws = (float*)d_ws;
  size_t off = 0;
  float* h1  = ws + off; off += (size_t)N_NODES * F_IN;   // 12.8M
  float* o1  = ws + off; off += (size_t)N_NODES * F_IN;   // 12.8M
  float* ex1 = ws + off; off += (size_t)E_TOT * H1;       // 3.4M
  float* h2  = ws + off; off += (size_t)N_NODES * C2;     // 3.2M
  float* as1 = ws + off; off += (size_t)N_NODES * H1;
  float* ad1 = ws + off; off += (size_t)N_NODES * H1;
  float* m1  = ws + off; off += (size_t)N_NODES * H1;
  float* dn1 = ws + off; off += (size_t)N_NODES * H1;
  float* as2 = ws + off; off += (size_t)N_NODES;
  float* ad2 = ws + off; off += (size_t)N_NODES;
  float* m2  = ws + off; off += (size_t)N_NODES;
  float* dn2 = ws + off; off += (size_t)N_NODES;
  float* ex2 = ws + off; off += (size_t)E_TOT;

  const float NEG_INF = -__builtin_inff();
  const int TPB = 256;

  // ---- layer 1 ----
  // h1 = x @ W1   (M=50000, N=256, K=256); one wave per 16x64 strip
  {
    long long waves = (long long)(N_NODES / 16) * (F_IN / 64);
    gemm_wmma_f32<<<cdiv(waves * 32, TPB), TPB, 0, stream>>>(x, W1, h1, N_NODES, F_IN, F_IN);
  }
  attn_coef<<<cdiv((long long)N_NODES * H1, TPB), TPB, 0, stream>>>(h1, aS1, aD1, as1, ad1, H1, C1);

  fill_f32<<<cdiv((long long)N_NODES * H1, TPB), TPB, 0, stream>>>(m1, NEG_INF, (size_t)N_NODES * H1);
  fill_f32<<<cdiv((long long)N_NODES * H1, TPB), TPB, 0, stream>>>(dn1, 0.f, (size_t)N_NODES * H1);
  fill_f32<<<cdiv((long long)N_NODES * F_IN, TPB), TPB, 0, stream>>>(o1, 0.f, (size_t)N_NODES * F_IN);

  edge_max_k<<<cdiv((long long)E_TOT * H1, TPB), TPB, 0, stream>>>(ei, as1, ad1, m1, H1);
  edge_exp_k<<<cdiv((long long)E_TOT * H1, TPB), TPB, 0, stream>>>(ei, as1, ad1, m1, ex1, dn1, H1);
  edge_aggr1<<<E_TOT, 256, 0, stream>>>(ei, h1, ex1, dn1, o1);

  bias_elu<<<cdiv((long long)N_NODES * F_IN, TPB), TPB, 0, stream>>>(o1, b1, (size_t)N_NODES * F_IN, F_IN);

  // ---- layer 2 ----
  // h2 = o1 @ W2   (M=50000, N=64, K=256)
  {
    long long waves = (long long)(N_NODES / 16) * (C2 / 64);
    gemm_wmma_f32<<<cdiv(waves * 32, TPB), TPB, 0, stream>>>(o1, W2, h2, N_NODES, C2, F_IN);
  }
  attn_coef<<<cdiv((long long)N_NODES, TPB), TPB, 0, stream>>>(h2, aS2, aD2, as2, ad2, 1, C2);

  fill_f32<<<cdiv((long long)N_NODES, TPB), TPB, 0, stream>>>(m2, NEG_INF, (size_t)N_NODES);
  fill_f32<<<cdiv((long long)N_NODES, TPB), TPB, 0, stream>>>(dn2, 0.f, (size_t)N_NODES);
  fill_f32<<<cdiv((long long)N_NODES * C2, TPB), TPB, 0, stream>>>(out, 0.f, (size_t)N_NODES * C2);

  edge_max_k<<<cdiv((long long)E_TOT, TPB), TPB, 0, stream>>>(ei, as2, ad2, m2, 1);
  edge_exp_k<<<cdiv((long long)E_TOT, TPB), TPB, 0, stream>>>(ei, as2, ad2, m2, ex2, dn2, 1);
  edge_aggr2<<<cdiv((long long)E_TOT * C2, 256), 256, 0, stream>>>(ei, h2, ex2, dn2, out);

  bias_add<<<cdiv((long long)N_NODES * C2, TPB), TPB, 0, stream>>>(out, b2, (size_t)N_NODES * C2, C2);
}